// GaussianAdapter_81166291959881
// MI455X (gfx1250) — compile-verified
//
#include <hip/hip_runtime.h>
#include <hip/hip_bf16.h>

#define BLOCK 256
#define DIN   34        // 7 + 3*9 raw floats per gaussian
#define DSH   9

typedef __attribute__((__vector_size__(16))) int v4i;
typedef __attribute__((address_space(1))) v4i* gv4_ptr;   // global
typedef __attribute__((address_space(3))) v4i* lv4_ptr;   // LDS

// 16-byte async copies through the CDNA5 async global<->LDS path (ASYNCcnt).
__device__ __forceinline__ void lds_async_load16(const void* g, void* l) {
    __builtin_amdgcn_global_load_async_to_lds_b128((gv4_ptr)g, (lv4_ptr)l, 0, 0);
}
__device__ __forceinline__ void lds_async_store16(void* g, void* l) {
    __builtin_amdgcn_global_store_async_from_lds_b128((gv4_ptr)g, (lv4_ptr)l, 0, 0);
}
__device__ __forceinline__ void wait_async0() {
    asm volatile("s_wait_asynccnt 0" ::: "memory");
}

__global__ __launch_bounds__(BLOCK) void gaussian_adapter_kernel(
    const float* __restrict__ means,   // (B,N,H,W,3)
    const float* __restrict__ opac,    // (B,N,H,W)
    const float* __restrict__ raw,     // (B,N,H,W,34)
    const float* __restrict__ poses,   // (B,N,4,4)
    float* __restrict__ out,
    int HW, int total)
{
    // LDS staging.  s_raw is reused for the SH output (27*256 floats fits in 34*256).
    __shared__ float s_raw[BLOCK * DIN];   // 34816 B : raw in  -> sh out
    __shared__ float s_mns[BLOCK * 3];     //  3072 B : means in -> global means out
    __shared__ float s_cov[BLOCK * 9];     //  9216 B
    __shared__ float s_scl[BLOCK * 3];     //  3072 B
    __shared__ float s_rot[BLOCK * 4];     //  4096 B
    __shared__ float s_pose[16];           //    64 B

    const int tid = threadIdx.x;
    const int p0  = blockIdx.x * BLOCK;    // block-uniform; 256 | H*W so (b,n) uniform
    const int p   = p0 + tid;

    if (tid < 16) {
        const int bn = p0 / HW;            // b*N + n
        s_pose[tid] = poses[bn * 16 + tid];
    }

    // ---- async stage: raw gaussians, 2176 x b128, contiguous + 16B aligned ----
    {
        const char* g = (const char*)(raw + (size_t)p0 * DIN);
        char*       l = (char*)s_raw;
#pragma unroll
        for (int i = 0; i < (BLOCK * DIN / 4 + BLOCK - 1) / BLOCK; ++i) {
            const int idx = i * BLOCK + tid;
            if (idx < BLOCK * DIN / 4)
                lds_async_load16(g + idx * 16, l + idx * 16);
        }
    }
    // ---- async stage: means, 192 x b128 ----
    {
        const char* g = (const char*)(means + (size_t)p0 * 3);
        char*       l = (char*)s_mns;
        if (tid < BLOCK * 3 / 4)
            lds_async_load16(g + tid * 16, l + tid * 16);
    }
    wait_async0();
    __syncthreads();

    // ---- pull my record into registers (LDS reads, stride 34 dwords: 2-way max) ----
    float r[DIN];
#pragma unroll
    for (int k = 0; k < DIN; ++k) r[k] = s_raw[tid * DIN + k];
    const float mx = s_mns[tid * 3 + 0];
    const float my = s_mns[tid * 3 + 1];
    const float mz = s_mns[tid * 3 + 2];
    float C[12];                            // rows 0..2 of the 4x4 pose
#pragma unroll
    for (int k = 0; k < 12; ++k) C[k] = s_pose[k];

    // scales: min(0.001*softplus(x), 0.3), softplus = max(x,0)+log1p(exp(-|x|))
    float scl[3];
#pragma unroll
    for (int k = 0; k < 3; ++k) {
        const float x  = r[k];
        const float sp = fmaxf(x, 0.0f) + log1pf(expf(-fabsf(x)));
        scl[k] = fminf(0.001f * sp, 0.3f);
    }

    // normalized quaternion (w,x,y,z)
    float qw = r[3], qx = r[4], qy = r[5], qz = r[6];
    {
        const float n   = sqrtf(qw*qw + qx*qx + qy*qy + qz*qz) + 1e-8f;
        const float inv = 1.0f / n;
        qw *= inv; qx *= inv; qy *= inv; qz *= inv;
    }

    // R(q), M = R*diag(s), L = M*M^T, G = C3 * L * C3^T
    float R[9];
    R[0] = 1.0f - 2.0f*(qy*qy + qz*qz);  R[1] = 2.0f*(qx*qy - qw*qz);  R[2] = 2.0f*(qx*qz + qw*qy);
    R[3] = 2.0f*(qx*qy + qw*qz);  R[4] = 1.0f - 2.0f*(qx*qx + qz*qz);  R[5] = 2.0f*(qy*qz - qw*qx);
    R[6] = 2.0f*(qx*qz - qw*qy);  R[7] = 2.0f*(qy*qz + qw*qx);  R[8] = 1.0f - 2.0f*(qx*qx + qy*qy);

    float M[9];
#pragma unroll
    for (int i = 0; i < 3; ++i)
#pragma unroll
        for (int k = 0; k < 3; ++k) M[i*3+k] = R[i*3+k] * scl[k];

    float L[9];
#pragma unroll
    for (int i = 0; i < 3; ++i)
#pragma unroll
        for (int j = 0; j < 3; ++j)
            L[i*3+j] = M[i*3+0]*M[j*3+0] + M[i*3+1]*M[j*3+1] + M[i*3+2]*M[j*3+2];

    float T[9];
#pragma unroll
    for (int i = 0; i < 3; ++i)
#pragma unroll
        for (int j = 0; j < 3; ++j)
            T[i*3+j] = C[i*4+0]*L[0*3+j] + C[i*4+1]*L[1*3+j] + C[i*4+2]*L[2*3+j];

    float G[9];
#pragma unroll
    for (int i = 0; i < 3; ++i)
#pragma unroll
        for (int l = 0; l < 3; ++l)
            G[i*3+l] = T[i*3+0]*C[l*4+0] + T[i*3+1]*C[l*4+1] + T[i*3+2]*C[l*4+2];

    // global means: rows of full 4x4 applied to homogeneous point
    float gm[3];
#pragma unroll
    for (int i = 0; i < 3; ++i)
        gm[i] = C[i*4+0]*mx + C[i*4+1]*my + C[i*4+2]*mz + C[i*4+3];

    // opacity: already perfectly coalesced, direct b32 load+store
    out[(size_t)total * 39 + p] = opac[p];

    // ---- stage outputs in LDS (odd dword strides -> bank-conflict-free) ----
    __syncthreads();                        // everyone done reading s_raw/s_mns
    const float msk[DSH] = {1.0f, 0.025f, 0.025f, 0.025f,
                            0.00625f, 0.00625f, 0.00625f, 0.00625f, 0.00625f};
#pragma unroll
    for (int c = 0; c < 3; ++c)
#pragma unroll
        for (int k = 0; k < DSH; ++k)
            s_raw[tid * 27 + c * DSH + k] = r[7 + c * DSH + k] * msk[k];
#pragma unroll
    for (int k = 0; k < 9; ++k) s_cov[tid * 9 + k] = G[k];
#pragma unroll
    for (int k = 0; k < 3; ++k) s_mns[tid * 3 + k] = gm[k];
#pragma unroll
    for (int k = 0; k < 3; ++k) s_scl[tid * 3 + k] = scl[k];
    s_rot[tid * 4 + 0] = qw; s_rot[tid * 4 + 1] = qx;
    s_rot[tid * 4 + 2] = qy; s_rot[tid * 4 + 3] = qz;
    __syncthreads();

    // ---- async store from LDS to the 5 contiguous output regions ----
    float* o_gm  = out;
    float* o_cov = out + (size_t)total * 3;
    float* o_sh  = out + (size_t)total * 12;
    float* o_scl = out + (size_t)total * 40;
    float* o_rot = out + (size_t)total * 43;

    {   // sh: 1728 x b128
        char* g = (char*)(o_sh + (size_t)p0 * 27);
        char* l = (char*)s_raw;
#pragma unroll
        for (int i = 0; i < 7; ++i) {
            const int idx = i * BLOCK + tid;
            if (idx < BLOCK * 27 / 4) lds_async_store16(g + idx * 16, l + idx * 16);
        }
    }
    {   // cov: 576 x b128
        char* g = (char*)(o_cov + (size_t)p0 * 9);
        char* l = (char*)s_cov;
#pragma unroll
        for (int i = 0; i < 3; ++i) {
            const int idx = i * BLOCK + tid;
            if (idx < BLOCK * 9 / 4) lds_async_store16(g + idx * 16, l + idx * 16);
        }
    }
    {   // global means: 192 x b128
        char* g = (char*)(o_gm + (size_t)p0 * 3);
        char* l = (char*)s_mns;
        if (tid < BLOCK * 3 / 4) lds_async_store16(g + tid * 16, l + tid * 16);
    }
    {   // scales: 192 x b128
        char* g = (char*)(o_scl + (size_t)p0 * 3);
        char* l = (char*)s_scl;
        if (tid < BLOCK * 3 / 4) lds_async_store16(g + tid * 16, l + tid * 16);
    }
    {   // rotations: 256 x b128
        char* g = (char*)(o_rot + (size_t)p0 * 4);
        char* l = (char*)s_rot;
        lds_async_store16(g + tid * 16, l + tid * 16);
    }
    // S_ENDPGM performs an implicit wait-idle (ASYNCcnt drains before retire).
}

extern "C" void kernel_launch(void* const* d_in, const int* in_sizes, int n_in,
                              void* d_out, int out_size, void* d_ws, size_t ws_size,
                              hipStream_t stream) {
    const float* means = (const float*)d_in[0];
    const float* opac  = (const float*)d_in[1];
    const float* raw   = (const float*)d_in[2];
    const float* poses = (const float*)d_in[3];
    float* out = (float*)d_out;

    const int total = in_sizes[1];          // B*N*H*W  (opacity count)
    const int BN    = in_sizes[3] / 16;     // B*N from (B,N,4,4)
    const int HW    = total / BN;

    const int blocks = total / BLOCK;       // total is a multiple of 256
    gaussian_adapter_kernel<<<blocks, BLOCK, 0, stream>>>(
        means, opac, raw, poses, out, HW, total);
}